// HaversineSmoothedLoss_44452911513880
// MI455X (gfx1250) — compile-verified
//
#include <hip/hip_runtime.h>

typedef __attribute__((ext_vector_type(2))) float v2f;
typedef __attribute__((ext_vector_type(8))) float v8f;

#define C_COLS   32768
#define C4       (C_COLS / 4)
#define THREADS  512
#define WAVES    (THREADS / 32)
#define D2R      0.017453292519943295f
#define NEG_2R_OVER_TAU (-12742.0f / 75.0f)        // -(2*6371)/75, folded at compile time
#define EPS_F    1e-9f
#define PI_F     3.14159265358979323846f
#define PIO2_F   1.57079632679489661923f

// odd Taylor/minimax sine, valid |x| <= pi/2, branch-free, err <= ~6e-8
__device__ __forceinline__ float sin_halfpi(float x) {
    float x2 = x * x;
    float p = -2.5050759689e-08f;
    p = __builtin_fmaf(p, x2, 2.7557314297e-06f);
    p = __builtin_fmaf(p, x2, -1.9841270114e-04f);
    p = __builtin_fmaf(p, x2, 8.3333337680e-03f);
    p = __builtin_fmaf(p, x2, -1.6666667163e-01f);
    return __builtin_fmaf(x * x2, p, x);
}

// sin(|x|) for |x| <= pi (fold sin(x)=sin(pi-x)); sign irrelevant (we square it)
__device__ __forceinline__ float sin_abs_pi(float x) {
    float ax = __builtin_fabsf(x);
    float f  = (ax > PIO2_F) ? (PI_F - ax) : ax;
    return sin_halfpi(f);
}

// asin(sqrt(a)) for a in [0,1], branch-free. Raw v_sqrt_f32 + one Cephes asin
// poly on z<=0.25, with the x>0.5 fold asin(x) = pi/2 - 2*asin(sqrt((1-x)/2)).
__device__ __forceinline__ float asin_sqrt(float a) {
    float x   = __builtin_amdgcn_sqrtf(a);         // x = sqrt(a) in [0,1]
    bool  big = a > 0.25f;                         // <=> x > 0.5
    float z   = big ? (1.0f - x) * 0.5f : a;       // z in [0, 0.25] either way
    float s   = big ? __builtin_amdgcn_sqrtf(z) : x;
    float p = 4.2163199048e-2f;
    p = __builtin_fmaf(p, z, 2.4181311049e-2f);
    p = __builtin_fmaf(p, z, 4.5470025998e-2f);
    p = __builtin_fmaf(p, z, 7.4953002686e-2f);
    p = __builtin_fmaf(p, z, 1.6666752422e-1f);
    float r = __builtin_fmaf(s * z, p, s);         // asin(s), s <= 0.5
    return big ? PIO2_F - 2.0f * r : r;
}

// g = {lat2r, lon2r, cos(lat2r), 0};  w = exp(-2R*asin(sqrt(a))/75)
__device__ __forceinline__ float hav_weight(float4 g, float lat1r, float lon1r,
                                            float cos_lat1) {
    float s1 = sin_halfpi(0.5f * (g.x - lat1r));
    float s2 = sin_abs_pi(0.5f * (g.y - lon1r));
    float a  = __builtin_fmaf(cos_lat1 * g.z, s2 * s2, s1 * s1);
    a = fminf(fmaxf(a, 0.0f), 1.0f);
    return __expf(asin_sqrt(a) * NEG_2R_OVER_TAU); // arg <= 0: no overflow path
}

// Sum the 16x16 f32 WMMA accumulator (B was all-ones -> every column holds the
// per-row totals). Per-lane vertical sum + cross-half add = full wave32 sum.
__device__ __forceinline__ float wmma_acc_total(v8f acc) {
    float t = acc[0] + acc[1] + acc[2] + acc[3] + acc[4] + acc[5] + acc[6] + acc[7];
    t += __shfl_xor(t, 16, 32);
    return t;
}

// ---- kernel 0: per-geocell precompute {lat2r, lon2r, cos(lat2r), 0} ----
__global__ __launch_bounds__(256)
void geo_precompute(const float* __restrict__ geo, float4* __restrict__ gtab) {
    int c = blockIdx.x * 256 + threadIdx.x;
    if (c < C_COLS) {
        float lat2r = geo[2 * c]     * D2R;
        float lon2r = geo[2 * c + 1] * D2R;
        gtab[c] = make_float4(lat2r, lon2r, cosf(lat2r), 0.0f);
    }
}

// ---- kernel 1: one block per row b ----
__global__ __launch_bounds__(THREADS)
void hav_loss_rows(const float* __restrict__ logits,
                   const float* __restrict__ latlon,
                   const float4* __restrict__ gtab,
                   float* __restrict__ row_loss) {
    __shared__ float4 wbuf[C4];                    // 128 KB: target weights w_c
    __shared__ float4 ebuf[C4];                    // 128 KB: e_c = exp(l_c - m)
    __shared__ float  sred[2 * WAVES];
    __shared__ float  sb[3];

    const int b    = blockIdx.x;
    const int tid  = threadIdx.x;
    const int lane = tid & 31;
    const int wave = tid >> 5;
    const float4* __restrict__ row4 = (const float4*)(logits + (size_t)b * C_COLS);

    const float lat1r    = latlon[2 * b]     * D2R;
    const float lon1r    = latlon[2 * b + 1] * D2R;
    const float cos_lat1 = cosf(lat1r);

    // ---------------- pass 1: row max (HBM pass, b128 loads) ----------------
    float m = -3.402823466e38f;
    for (int j = tid; j < C4; j += THREADS) {
        float4 l = row4[j];
        m = fmaxf(m, fmaxf(fmaxf(l.x, l.y), fmaxf(l.z, l.w)));
    }
    for (int off = 16; off > 0; off >>= 1)
        m = fmaxf(m, __shfl_xor(m, off, 32));
    if (lane == 0) sred[wave] = m;
    __syncthreads();
    if (tid == 0) {
        float mm = sred[0];
        for (int i = 1; i < WAVES; ++i) mm = fmaxf(mm, sred[i]);
        sb[0] = mm;
    }
    __syncthreads();
    m = sb[0];

    // ------- pass 2: sumW, sumExp (L2 pass); stash w_c, e_c in LDS (b128) -------
    v8f accW = {0.f, 0.f, 0.f, 0.f, 0.f, 0.f, 0.f, 0.f};
    v8f accE = {0.f, 0.f, 0.f, 0.f, 0.f, 0.f, 0.f, 0.f};
    const v2f ones = {1.0f, 1.0f};
    for (int j = tid; j < C4; j += THREADS) {
        float4 l = row4[j];
        float e0 = __expf(l.x - m), e1 = __expf(l.y - m);
        float e2 = __expf(l.z - m), e3 = __expf(l.w - m);

        const int cb = 4 * j;
        float w0 = hav_weight(gtab[cb + 0], lat1r, lon1r, cos_lat1);
        float w1 = hav_weight(gtab[cb + 1], lat1r, lon1r, cos_lat1);
        float w2 = hav_weight(gtab[cb + 2], lat1r, lon1r, cos_lat1);
        float w3 = hav_weight(gtab[cb + 3], lat1r, lon1r, cos_lat1);

        wbuf[j] = make_float4(w0, w1, w2, w3);
        ebuf[j] = make_float4(e0, e1, e2, e3);

        v2f aW0 = {w0, w1}, aW1 = {w2, w3};
        accW = __builtin_amdgcn_wmma_f32_16x16x4_f32(false, aW0, false, ones,
                                                     (short)0, accW, false, false);
        accW = __builtin_amdgcn_wmma_f32_16x16x4_f32(false, aW1, false, ones,
                                                     (short)0, accW, false, false);
        v2f aE0 = {e0, e1}, aE1 = {e2, e3};
        accE = __builtin_amdgcn_wmma_f32_16x16x4_f32(false, aE0, false, ones,
                                                     (short)0, accE, false, false);
        accE = __builtin_amdgcn_wmma_f32_16x16x4_f32(false, aE1, false, ones,
                                                     (short)0, accE, false, false);
    }
    float tW = wmma_acc_total(accW);
    float tE = wmma_acc_total(accE);
    if (lane == 0) { sred[wave] = tW; sred[WAVES + wave] = tE; }
    __syncthreads();
    if (tid == 0) {
        float sw = 0.f, se = 0.f;
        for (int i = 0; i < WAVES; ++i) { sw += sred[i]; se += sred[WAVES + i]; }
        sb[1] = se; sb[2] = sw;
    }
    __syncthreads();
    const float invE = 1.0f / sb[1];
    const float Wsum = sb[2];

    // ------- pass 3: loss = sum_c w_c * -log(e_c/Esum + eps)  (LDS only) -------
    v8f accL = {0.f, 0.f, 0.f, 0.f, 0.f, 0.f, 0.f, 0.f};
    for (int j = tid; j < C4; j += THREADS) {
        float4 w = wbuf[j];
        float4 e = ebuf[j];
        float t0 = -__logf(__builtin_fmaf(e.x, invE, EPS_F));
        float t1 = -__logf(__builtin_fmaf(e.y, invE, EPS_F));
        float t2 = -__logf(__builtin_fmaf(e.z, invE, EPS_F));
        float t3 = -__logf(__builtin_fmaf(e.w, invE, EPS_F));
        v2f aL0 = {w.x * t0, w.y * t1};
        v2f aL1 = {w.z * t2, w.w * t3};
        accL = __builtin_amdgcn_wmma_f32_16x16x4_f32(false, aL0, false, ones,
                                                     (short)0, accL, false, false);
        accL = __builtin_amdgcn_wmma_f32_16x16x4_f32(false, aL1, false, ones,
                                                     (short)0, accL, false, false);
    }
    float tL = wmma_acc_total(accL);
    if (lane == 0) sred[wave] = tL;
    __syncthreads();
    if (tid == 0) {
        float s = 0.f;
        for (int i = 0; i < WAVES; ++i) s += sred[i];
        row_loss[b] = s / Wsum;                    // target row-normalization
    }
}

// ---- kernel 2: deterministic mean over 512 rows ----
__global__ __launch_bounds__(512)
void reduce_mean_512(const float* __restrict__ row_loss, float* __restrict__ out) {
    __shared__ float sred[16];
    const int tid  = threadIdx.x;
    const int lane = tid & 31;
    const int wave = tid >> 5;
    float v = row_loss[tid];
    for (int off = 16; off > 0; off >>= 1)
        v += __shfl_xor(v, off, 32);
    if (lane == 0) sred[wave] = v;
    __syncthreads();
    if (tid == 0) {
        float s = 0.f;
        for (int i = 0; i < 16; ++i) s += sred[i];
        out[0] = s / 512.0f;
    }
}

extern "C" void kernel_launch(void* const* d_in, const int* in_sizes, int n_in,
                              void* d_out, int out_size, void* d_ws, size_t ws_size,
                              hipStream_t stream) {
    const float* logits = (const float*)d_in[0];   // (B, C) f32
    const float* latlon = (const float*)d_in[1];   // (B, 2) f32
    const float* geo    = (const float*)d_in[2];   // (C, 2) f32
    float* out = (float*)d_out;

    // d_ws layout: [0 .. B) row_loss floats, then float4 gtab[C] (16B aligned)
    float*  row_loss = (float*)d_ws;
    float4* gtab     = (float4*)((char*)d_ws + 512 * sizeof(float));

    const int B = in_sizes[1] / 2;                 // 512
    geo_precompute<<<(C_COLS + 255) / 256, 256, 0, stream>>>(geo, gtab);
    hav_loss_rows<<<B, THREADS, 0, stream>>>(logits, latlon, gtab, row_loss);
    reduce_mean_512<<<1, 512, 0, stream>>>(row_loss, out);
}